// MeshGNN_77627238908603
// MI455X (gfx1250) — compile-verified
//
#include <hip/hip_runtime.h>
#include <math.h>

#define NN   60000
#define HH   128
#define KK   15
#define OUTD 259            // 64 (q) + 64 (k) + 131 (px)
#define YLD  272            // Y leading dim, padded to 17 * 16
#define WROWS 320           // W_all rows padded to 20 * 16 (zero-filled tail)
#define NTILES_N 17         // valid 16-col tiles in Y

typedef __attribute__((ext_vector_type(2))) float v2f;
typedef __attribute__((ext_vector_type(8))) float v8f;

// ---------------------------------------------------------------------------
// Kernel 1: pack fused weight matrix W_all[WROWS][128] and bias b_all[WROWS].
//   rows [0,64)    = W_q                     bias = b_q
//   rows [64,128)  = W_k                     bias = b_k
//   rows [128,259) = W_concat[:, 0:128]      bias = b_concat  (baked into px)
//   rows [259,320) = 0                       bias = 0
// ---------------------------------------------------------------------------
__global__ void mesh_pack_weights(const float* __restrict__ Wq,
                                  const float* __restrict__ Wk,
                                  const float* __restrict__ Wc,
                                  const float* __restrict__ bq,
                                  const float* __restrict__ bk,
                                  const float* __restrict__ bc,
                                  float* __restrict__ W_all,
                                  float* __restrict__ b_all) {
    int idx = blockIdx.x * blockDim.x + threadIdx.x;
    if (idx < WROWS * HH) {
        int r = idx >> 7;      // row (output feature)
        int c = idx & 127;     // input channel
        float v = 0.0f;
        if      (r < 64)   v = Wq[r * HH + c];
        else if (r < 128)  v = Wk[(r - 64) * HH + c];
        else if (r < OUTD) v = Wc[(r - 128) * 131 + c];   // first 128 cols of W_concat row
        W_all[idx] = v;
    }
    if (idx < WROWS) {
        float v = 0.0f;
        if      (idx < 64)   v = bq[idx];
        else if (idx < 128)  v = bk[idx - 64];
        else if (idx < OUTD) v = bc[idx - 128];
        b_all[idx] = v;
    }
}

// ---------------------------------------------------------------------------
// Kernel 2: Y[N, YLD] = X[N,128] @ W_all.T + b_all   via V_WMMA_F32_16X16X4_F32
// One wave per 32x64 output block: 2 row-tiles x 4 col-tiles = 8 accumulators.
// Per K=4 step: 2 A float2 loads + 4 B float2 loads -> 8 WMMAs (high reuse).
// A operand (16x4 f32): lane l -> M = l&15, K pair = (l>>4)*2  => float2 load.
// B operand (4x16 f32): lane l -> N = l&15, K pair = (l>>4)*2  => float2 load.
// D (16x16 f32, 8 VGPRs): VGPR v -> M = v + 8*(l>>4), N = l&15.
// ---------------------------------------------------------------------------
__global__ void mesh_gemm_wmma(const float* __restrict__ X,
                               const float* __restrict__ W_all,
                               const float* __restrict__ b_all,
                               float* __restrict__ Y) {
    const int lane    = threadIdx.x;            // 0..31, wave32
    const int rowBase = blockIdx.x * 32;        // 1875 blocks * 32 = 60000 rows
    const int colBase = blockIdx.y * 64;        // 5 groups * 64 = 320 padded cols
    const int r    = lane & 15;
    const int half = lane >> 4;
    const int ncol = lane & 15;

    const float* arow0 = X + (size_t)(rowBase + r) * HH + 2 * half;
    const float* arow1 = arow0 + (size_t)16 * HH;
    const float* brow  = W_all + (size_t)(colBase + ncol) * HH + 2 * half;

    v8f acc[2][4];
#pragma unroll
    for (int ct = 0; ct < 4; ++ct) {
        const float bias = b_all[colBase + ct * 16 + ncol];  // bias depends on N only
#pragma unroll
        for (int v = 0; v < 8; ++v) { acc[0][ct][v] = bias; acc[1][ct][v] = bias; }
    }

#pragma unroll
    for (int s = 0; s < HH / 4; ++s) {
        v2f a0 = *(const v2f*)(arow0 + 4 * s);
        v2f a1 = *(const v2f*)(arow1 + 4 * s);
        v2f b[4];
#pragma unroll
        for (int ct = 0; ct < 4; ++ct)
            b[ct] = *(const v2f*)(brow + (size_t)ct * 16 * HH + 4 * s);
#pragma unroll
        for (int ct = 0; ct < 4; ++ct) {
            acc[0][ct] = __builtin_amdgcn_wmma_f32_16x16x4_f32(
                false, a0, false, b[ct], (short)0, acc[0][ct], false, false);
            acc[1][ct] = __builtin_amdgcn_wmma_f32_16x16x4_f32(
                false, a1, false, b[ct], (short)0, acc[1][ct], false, false);
        }
    }

#pragma unroll
    for (int ct = 0; ct < 4; ++ct) {
        const int tt = blockIdx.y * 4 + ct;     // global 16-col tile index
        if (tt < NTILES_N) {                    // uniform per wave: skip pad tiles
            float* y0 = Y + (size_t)rowBase * YLD + tt * 16;
#pragma unroll
            for (int v = 0; v < 8; ++v) {
                y0[(size_t)(v + 8 * half) * YLD + ncol]      = acc[0][ct][v];
                y0[(size_t)(16 + v + 8 * half) * YLD + ncol] = acc[1][ct][v];
            }
        }
    }
}

// ---------------------------------------------------------------------------
// Kernel 3: per-node attention + aggregation + output head.
// One wave per node, 8 waves (256 threads) per block.
//   Y row layout: [0,64) = q (incl b_q), [64,128) = kx (incl b_k),
//                 [128,259) = px = x@Wc_x.T + b_concat
// ---------------------------------------------------------------------------
__global__ void mesh_edge_kernel(const float* __restrict__ points,
                                 const int*   __restrict__ edges,   // [2, N*K]
                                 const float* __restrict__ Y,
                                 const float* __restrict__ Wc,      // [131,131]
                                 const float* __restrict__ Wout,    // [3,131]
                                 const float* __restrict__ bout,    // [3]
                                 float* __restrict__ out,
                                 float inv_scale) {
    const int wave = threadIdx.x >> 5;
    const int lane = threadIdx.x & 31;
    const int n    = blockIdx.x * 8 + wave;   // grid = 7500 -> exactly N nodes

    __shared__ float s_attn[8][16];
    __shared__ int   s_j[8][16];

    // ---- phase 1: lane kk < K computes masked logit for neighbor kk ----
    float logit = -INFINITY;
    int   j     = 0;
    if (lane < KK) {
        j = edges[(size_t)NN * KK + (size_t)n * KK + lane];  // dst row
        const float* qrow = Y + (size_t)n * YLD;             // broadcast across lanes
        const float* krow = Y + (size_t)j * YLD + 64;
        float acc = 0.0f;
#pragma unroll 8
        for (int c = 0; c < 64; ++c) acc = fmaf(qrow[c], krow[c], acc);
        float mask = (j != 0) ? 1.0f : 0.0f;                 // ref: softmax(scores*mask)
        logit = acc * inv_scale * mask;
    }
    // softmax across lanes 0..14 (lanes >= K contribute exp = 0)
    float m = logit;
#pragma unroll
    for (int off = 16; off; off >>= 1) m = fmaxf(m, __shfl_xor(m, off, 32));
    float e = (lane < KK) ? __expf(logit - m) : 0.0f;
    float ssum = e;
#pragma unroll
    for (int off = 16; off; off >>= 1) ssum += __shfl_xor(ssum, off, 32);
    float attn = e / ssum;

    if (lane < KK) { s_attn[wave][lane] = attn; s_j[wave][lane] = j; }
    __syncthreads();

    float attns[KK];
    int   js[KK];
#pragma unroll
    for (int kk = 0; kk < KK; ++kk) { attns[kk] = s_attn[wave][kk]; js[kk] = s_j[wave][kk]; }

    // ---- attention-weighted relative position (3 dims, redundant per lane) ----
    const float p0 = points[(size_t)n * 3 + 0];
    const float p1 = points[(size_t)n * 3 + 1];
    const float p2 = points[(size_t)n * 3 + 2];
    float ag0 = 0.0f, ag1 = 0.0f, ag2 = 0.0f;
#pragma unroll
    for (int kk = 0; kk < KK; ++kk) {
        const float* pj = points + (size_t)js[kk] * 3;
        ag0 = fmaf(attns[kk], pj[0] - p0, ag0);
        ag1 = fmaf(attns[kk], pj[1] - p1, ag1);
        ag2 = fmaf(attns[kk], pj[2] - p2, ag2);
    }

    // ---- feature aggregation + output head, lanes stride over 131 dims ----
    float part0 = 0.0f, part1 = 0.0f, part2 = 0.0f;
    for (int d = lane; d < 131; d += 32) {
        float acc = 0.0f;
#pragma unroll
        for (int kk = 0; kk < KK; ++kk)
            acc = fmaf(attns[kk], Y[(size_t)js[kk] * YLD + 128 + d], acc);
        // + (sum_k attn * rel_pos) @ Wc_pos.T  (cols 128..130 of W_concat row d)
        const float* wrow = Wc + (size_t)d * 131 + 128;
        acc = fmaf(ag0, wrow[0], acc);
        acc = fmaf(ag1, wrow[1], acc);
        acc = fmaf(ag2, wrow[2], acc);
        part0 = fmaf(Wout[0 * 131 + d], acc, part0);
        part1 = fmaf(Wout[1 * 131 + d], acc, part1);
        part2 = fmaf(Wout[2 * 131 + d], acc, part2);
    }
#pragma unroll
    for (int off = 16; off; off >>= 1) {
        part0 += __shfl_xor(part0, off, 32);
        part1 += __shfl_xor(part1, off, 32);
        part2 += __shfl_xor(part2, off, 32);
    }
    if (lane == 0) {
        out[(size_t)n * 3 + 0] = p0 + part0 + bout[0];
        out[(size_t)n * 3 + 1] = p1 + part1 + bout[1];
        out[(size_t)n * 3 + 2] = p2 + part2 + bout[2];
    }
}

// ---------------------------------------------------------------------------
extern "C" void kernel_launch(void* const* d_in, const int* in_sizes, int n_in,
                              void* d_out, int out_size, void* d_ws, size_t ws_size,
                              hipStream_t stream) {
    const float* points = (const float*)d_in[0];   // [N,3]
    const float* x      = (const float*)d_in[1];   // [N,128]
    const int*   edges  = (const int*)d_in[2];     // [2, N*K]
    const float* Wc     = (const float*)d_in[3];   // [131,131]
    const float* bc     = (const float*)d_in[4];   // [131]
    const float* Wout   = (const float*)d_in[5];   // [3,131]
    const float* bout   = (const float*)d_in[6];   // [3]
    const float* Wq     = (const float*)d_in[7];   // [64,128]
    const float* bq     = (const float*)d_in[8];   // [64]
    const float* Wk     = (const float*)d_in[9];   // [64,128]
    const float* bk     = (const float*)d_in[10];  // [64]
    float* out = (float*)d_out;

    // workspace layout (bytes)
    char* ws = (char*)d_ws;
    float* W_all = (float*)(ws);                   // 320*128*4 = 163840 B
    float* b_all = (float*)(ws + 163840);          // 320*4     = 1280 B
    float* Y     = (float*)(ws + 165376);          // 60000*272*4 = 65.28 MB

    (void)in_sizes; (void)n_in; (void)out_size; (void)ws_size;

    // 1) pack fused weights (zero-padded to 320 rows)
    mesh_pack_weights<<<(WROWS * HH + 255) / 256, 256, 0, stream>>>(
        Wq, Wk, Wc, bq, bk, bc, W_all, b_all);

    // 2) fused node GEMM via f32 WMMA: Y = X @ W_all.T + b_all
    dim3 gg(NN / 32, 5);   // 1875 row blocks x 5 col groups (64 cols each)
    mesh_gemm_wmma<<<gg, 32, 0, stream>>>(x, W_all, b_all, Y);

    // 3) attention + aggregation + output head
    const float inv_scale = 1.0f / (sqrtf(64.0f) + 1e-6f);
    mesh_edge_kernel<<<NN / 8, 256, 0, stream>>>(
        points, edges, Y, Wc, Wout, bout, out, inv_scale);
}